// GlobalAttentionPooling_49289044689302
// MI455X (gfx1250) — compile-verified
//
#include <hip/hip_runtime.h>
#include <hip/hip_bf16.h>
#include <cstdint>

#define N_NODES    500000
#define DIM        256
#define NSEG       2048
#define CACHE_ROWS 64        // rows of feat pinned in LDS per segment (64 KB)
#define MAX_LGATE  1024      // gate/e values kept in LDS (4 KB); ws fallback beyond
#define SMEM_FLOATS (CACHE_ROWS * DIM + MAX_LGATE + 8 + 2)

typedef __attribute__((ext_vector_type(2))) float v2f;
typedef __attribute__((ext_vector_type(8))) float v8f;

// ------------- fused per-segment kernel: gate -> softmax -> weighted sum -------------
// One block per segment (segment_ids sorted => contiguous row range).
// No atomics; feat slab is read once from HBM, re-read from LDS (first 64 rows) / L2.
__global__ __launch_bounds__(256) void k_fused(const float* __restrict__ feat,
                                               const long long* __restrict__ seg,
                                               const float* __restrict__ Wg,
                                               const float* __restrict__ bg,
                                               float* __restrict__ gate_ws,   // N floats (overflow spill)
                                               float* __restrict__ wfeat,     // B*D
                                               float* __restrict__ segflag) { // B (1=non-empty)
    extern __shared__ float smem[];
    float* lrow   = smem;                          // CACHE_ROWS*DIM
    float* lgate  = lrow + CACHE_ROWS * DIM;       // MAX_LGATE
    float* red    = lgate + MAX_LGATE;             // 8
    float* s_stat = red + 8;                       // 2: [0]=max, [1]=1/sum

    const int b    = blockIdx.x;
    const int tid  = threadIdx.x;
    const int wv   = tid >> 5;
    const int lane = tid & 31;

    // ---- binary search segment bounds ----
    long long v = (long long)b;
    int lo = 0, hi = N_NODES;
    while (lo < hi) { int mid = (lo + hi) >> 1; if (seg[mid] < v) lo = mid + 1; else hi = mid; }
    const int start = lo;
    v = (long long)b + 1; hi = N_NODES;
    while (lo < hi) { int mid = (lo + hi) >> 1; if (seg[mid] < v) lo = mid + 1; else hi = mid; }
    const int end = lo;
    const int cnt = end - start;

    if (cnt == 0) {                       // empty segment: zero row, no bias
        wfeat[(size_t)b * DIM + tid] = 0.0f;
        if (tid == 0) segflag[b] = 0.0f;
        return;
    }

    // ---- Phase A: gate dot products; cache rows in LDS while in registers ----
    const float4* wg4 = (const float4*)Wg;
    const float4  w0 = wg4[lane * 2], w1 = wg4[lane * 2 + 1];
    const float   bgv = bg[0];
    float wmaxv = -INFINITY;
    for (int i = wv; i < cnt; i += 8) {
        const int r = start + i;
        const float4* fr = (const float4*)(feat + (size_t)r * DIM);
        float4 a0 = fr[lane * 2], a1 = fr[lane * 2 + 1];
        if (i < CACHE_ROWS) {
            float4* dst = (float4*)(lrow + i * DIM);
            dst[lane * 2] = a0; dst[lane * 2 + 1] = a1;
        }
        float s = a0.x * w0.x + a0.y * w0.y + a0.z * w0.z + a0.w * w0.w
                + a1.x * w1.x + a1.y * w1.y + a1.z * w1.z + a1.w * w1.w;
        #pragma unroll
        for (int off = 16; off > 0; off >>= 1) s += __shfl_xor(s, off, 32);
        float g = s + bgv;
        if (lane == 0) { if (i < MAX_LGATE) lgate[i] = g; else gate_ws[r] = g; }
        wmaxv = fmaxf(wmaxv, g);
    }
    if (lane == 0) red[wv] = wmaxv;
    __syncthreads();
    if (tid == 0) {
        float m = -INFINITY;
        #pragma unroll
        for (int j = 0; j < 8; ++j) m = fmaxf(m, red[j]);
        s_stat[0] = m;
    }
    __syncthreads();
    const float bmax = s_stat[0];

    // ---- Phase A2: e = exp(g - max); block sum ----
    float psum = 0.0f;
    for (int i = tid; i < cnt; i += 256) {
        float g = (i < MAX_LGATE) ? lgate[i] : gate_ws[start + i];
        float e = __expf(g - bmax);
        if (i < MAX_LGATE) lgate[i] = e; else gate_ws[start + i] = e;
        psum += e;
    }
    #pragma unroll
    for (int off = 16; off > 0; off >>= 1) psum += __shfl_xor(psum, off, 32);
    __syncthreads();                 // red[] reuse + lgate writes visible
    if (lane == 0) red[wv] = psum;
    __syncthreads();
    if (tid == 0) {
        float s = 0.0f;
        #pragma unroll
        for (int j = 0; j < 8; ++j) s += red[j];
        s_stat[1] = 1.0f / s;        // cnt>0 => s>0
    }
    __syncthreads();
    const float inv = s_stat[1];

    // ---- Phase B: weighted column sum; thread t owns feature column t ----
    auto wgt = [&](int i) -> float {
        return (i < MAX_LGATE) ? lgate[i] : gate_ws[start + i];
    };
    float acc = 0.0f;
    const int cached = (cnt < CACHE_ROWS) ? cnt : CACHE_ROWS;
    int i = 0;
    for (; i + 4 <= cached; i += 4) {
        acc += lgate[i]     * lrow[(i)     * DIM + tid]
             + lgate[i + 1] * lrow[(i + 1) * DIM + tid]
             + lgate[i + 2] * lrow[(i + 2) * DIM + tid]
             + lgate[i + 3] * lrow[(i + 3) * DIM + tid];
    }
    for (; i < cached; ++i) acc += lgate[i] * lrow[i * DIM + tid];
    for (; i + 4 <= cnt; i += 4) {
        __builtin_prefetch(feat + (size_t)(start + i + 4) * DIM + tid, 0, 0);
        acc += wgt(i)     * feat[(size_t)(start + i)     * DIM + tid]
             + wgt(i + 1) * feat[(size_t)(start + i + 1) * DIM + tid]
             + wgt(i + 2) * feat[(size_t)(start + i + 2) * DIM + tid]
             + wgt(i + 3) * feat[(size_t)(start + i + 3) * DIM + tid];
    }
    for (; i < cnt; ++i) acc += wgt(i) * feat[(size_t)(start + i) * DIM + tid];

    wfeat[(size_t)b * DIM + tid] = acc * inv;
    if (tid == 0) segflag[b] = 1.0f;
}

// ------------- K4: readout = wfeat @ Wf + flag*bf  (exact fp32 WMMA) -------------
// One wave32 per 16x16 tile; V_WMMA_F32_16X16X4_F32, layouts per ISA 7.12.2.
__global__ __launch_bounds__(256) void k4_gemm(const float* __restrict__ wfeat,
                                               const float* __restrict__ Wf,
                                               const float* __restrict__ bf,
                                               const float* __restrict__ segflag,
                                               float* __restrict__ out) {
    int wave = (blockIdx.x * blockDim.x + threadIdx.x) >> 5;  // 0..2047
    int lane = threadIdx.x & 31;
    int tileM = wave >> 4;   // 0..127 (segment tiles)
    int tileN = wave & 15;   // 0..15  (feature tiles)

    int mn = lane & 15;            // A: row M ; B: col N
    int kk = (lane >> 4) << 1;     // lanes 0-15 -> K+0/K+1, lanes 16-31 -> K+2/K+3

    const float* arow = wfeat + (size_t)(tileM * 16 + mn) * DIM;
    const float* bcol = Wf + tileN * 16 + mn;

    v8f c = {};
    #pragma unroll 4
    for (int k = 0; k < DIM; k += 4) {
        v2f a, bm;
        a[0]  = arow[k + kk];
        a[1]  = arow[k + kk + 1];
        bm[0] = bcol[(size_t)(k + kk) * DIM];
        bm[1] = bcol[(size_t)(k + kk + 1) * DIM];
        c = __builtin_amdgcn_wmma_f32_16x16x4_f32(
                false, a, false, bm, (short)0, c, false, false);
    }

    int col = tileN * 16 + mn;
    float bfv = bf[col];
    int rbase = tileM * 16 + ((lane >> 4) << 3);
    #pragma unroll
    for (int j = 0; j < 8; ++j) {
        int row = rbase + j;
        out[(size_t)row * DIM + col] = c[j] + segflag[row] * bfv;
    }
}

extern "C" void kernel_launch(void* const* d_in, const int* in_sizes, int n_in,
                              void* d_out, int out_size, void* d_ws, size_t ws_size,
                              hipStream_t stream) {
    const float*     feat = (const float*)d_in[0];
    const long long* seg  = (const long long*)d_in[1];  // int64 per reference
    const float*     Wg   = (const float*)d_in[2];
    const float*     bg   = (const float*)d_in[3];
    const float*     Wf   = (const float*)d_in[4];
    const float*     bf   = (const float*)d_in[5];
    float*           out  = (float*)d_out;

    // workspace: gate spill | wfeat | segflag   (~4.1 MB)
    float* gate_ws = (float*)d_ws;                   // N floats
    float* wfeat   = gate_ws + N_NODES;              // B*D floats
    float* segflag = wfeat + (size_t)NSEG * DIM;     // B floats

    const size_t smem_bytes = SMEM_FLOATS * sizeof(float);  // ~69.7 KB dynamic LDS
    k_fused<<<NSEG, 256, smem_bytes, stream>>>(feat, seg, Wg, bg, gate_ws, wfeat, segflag);

    // (B/16)*(D/16) = 2048 tiles, 8 waves per 256-thread block -> 256 blocks, full waves
    k4_gemm<<<(NSEG / 16) * (DIM / 16) / 8, 256, 0, stream>>>(wfeat, Wf, bf, segflag, out);
}